// DBMLLoss_79328045957883
// MI455X (gfx1250) — compile-verified
//
#include <hip/hip_runtime.h>

// DBML loss on MI455X (gfx1250, wave32).
// Phase 1: sim = feats @ feats^T via V_WMMA_F32_16X16X4_F32, 2x2 register-blocked
//          per wave (4 loads -> 4 WMMAs) so the L1 return path isn't the limiter.
// Phase 2: fused row-pass (all row stats + adaptive selection + per-row loss),
//          wave32 shuffle reductions + one 8-entry LDS stage (3 barriers total).
// Phase 3: deterministic single-block final reduction.

typedef float v2f __attribute__((ext_vector_type(2)));
typedef float v8f __attribute__((ext_vector_type(8)));

#define BN 4096   // batch (rows/cols of sim)
#define DK 512    // feature dim

#define POS_A 1.0f
#define POS_B 0.5f
#define NEG_A 0.6f
#define NEG_B 0.5f
#define MARGIN 0.1f
#define WEIGHT 0.5f
#define EPSV 1e-5f

// ---------------------------------------------------------------------------
// Kernel 1: fp32 WMMA GEMM, 2x2 register blocking. 8 waves/block arranged
// 2(M)x4(N); each wave owns a 32x32 C region (4 accumulators) -> block covers
// 64x128. A and B fragments load with the identical row-major float2 pattern
// since B = feats^T, and each fragment is reused by two WMMAs.
// ---------------------------------------------------------------------------
__global__ __launch_bounds__(256)
void dbml_gemm_wmma(const float* __restrict__ feats, float* __restrict__ sim) {
  const int lane = threadIdx.x & 31;
  const int wave = threadIdx.x >> 5;
  const int wm = wave >> 2;      // 0..1
  const int wn = wave & 3;       // 0..3
  const int l16 = lane & 15;
  const int kh  = lane >> 4;     // half-wave selects K pair {0,1} vs {2,3}

  const int m0 = blockIdx.y * 64 + wm * 32;
  const int n0 = blockIdx.x * 128 + wn * 32;

  const float* a0p = feats + (size_t)(m0 + l16) * DK + 2 * kh;
  const float* a1p = a0p + (size_t)16 * DK;
  const float* b0p = feats + (size_t)(n0 + l16) * DK + 2 * kh;
  const float* b1p = b0p + (size_t)16 * DK;

  v8f acc00 = {}, acc01 = {}, acc10 = {}, acc11 = {};
#pragma unroll 2
  for (int k = 0; k < DK; k += 4) {
    v2f a0 = *(const v2f*)(a0p + k);
    v2f a1 = *(const v2f*)(a1p + k);
    v2f b0 = *(const v2f*)(b0p + k);
    v2f b1 = *(const v2f*)(b1p + k);
    acc00 = __builtin_amdgcn_wmma_f32_16x16x4_f32(false, a0, false, b0,
                                                  (short)0, acc00, false, false);
    acc01 = __builtin_amdgcn_wmma_f32_16x16x4_f32(false, a0, false, b1,
                                                  (short)0, acc01, false, false);
    acc10 = __builtin_amdgcn_wmma_f32_16x16x4_f32(false, a1, false, b0,
                                                  (short)0, acc10, false, false);
    acc11 = __builtin_amdgcn_wmma_f32_16x16x4_f32(false, a1, false, b1,
                                                  (short)0, acc11, false, false);
  }

  // C layout per tile: VGPR i, lanes 0-15 -> M=+i, lanes 16-31 -> M=+i+8; N=+l16
  {
    float* o = sim + (size_t)(m0 + 8 * kh) * BN + n0 + l16;
#pragma unroll
    for (int i = 0; i < 8; ++i) o[(size_t)i * BN] = acc00[i];
  }
  {
    float* o = sim + (size_t)(m0 + 8 * kh) * BN + (n0 + 16) + l16;
#pragma unroll
    for (int i = 0; i < 8; ++i) o[(size_t)i * BN] = acc01[i];
  }
  {
    float* o = sim + (size_t)(m0 + 16 + 8 * kh) * BN + n0 + l16;
#pragma unroll
    for (int i = 0; i < 8; ++i) o[(size_t)i * BN] = acc10[i];
  }
  {
    float* o = sim + (size_t)(m0 + 16 + 8 * kh) * BN + (n0 + 16) + l16;
#pragma unroll
    for (int i = 0; i < 8; ++i) o[(size_t)i * BN] = acc11[i];
  }
}

// ---- wave32 shuffle reductions (no LDS, no barriers) ----------------------
__device__ __forceinline__ float waveSum(float v) {
#pragma unroll
  for (int m = 16; m >= 1; m >>= 1) v += __shfl_xor(v, m, 32);
  return v;
}
__device__ __forceinline__ float waveMin(float v) {
#pragma unroll
  for (int m = 16; m >= 1; m >>= 1) v = fminf(v, __shfl_xor(v, m, 32));
  return v;
}
__device__ __forceinline__ float waveMax(float v) {
#pragma unroll
  for (int m = 16; m >= 1; m >>= 1) v = fmaxf(v, __shfl_xor(v, m, 32));
  return v;
}

// ---------------------------------------------------------------------------
// Kernel 2: fused per-row pass. One 256-thread block (8 waves) per row; each
// thread covers 16 columns, caching sim values + same-label bits in registers
// so the post-barrier selection phase needs no global re-read.
// ---------------------------------------------------------------------------
__global__ __launch_bounds__(256)
void dbml_rowpass(const float* __restrict__ sim, const int* __restrict__ labels,
                  float* __restrict__ lossrow) {
  const int r = blockIdx.x;
  const int tid = threadIdx.x;
  const int lane = tid & 31;
  const int wave = tid >> 5;
  const int myl = labels[r];
  const float* row = sim + (size_t)r * BN;

  __shared__ float xbuf[6][8];

  // ---- phase 1: full-row stats ----
  float sims[16];
  unsigned samemask = 0;
  float s = 0.f, sq = 0.f, mnp = 3.0e38f, mxn = -3.0e38f;
  float pc = 0.f, nc = 0.f;
#pragma unroll
  for (int i = 0; i < 16; ++i) {
    const int c = tid + i * 256;
    const float v = row[c];
    sims[i] = v;
    const bool same = (labels[c] == myl);
    samemask |= (same ? 1u : 0u) << i;
    s += v;
    sq += v * v;
    const bool pos = same && (v < 1.0f - EPSV);
    if (pos) { mnp = fminf(mnp, v); pc += 1.f; }
    if (!same) { mxn = fmaxf(mxn, v); nc += 1.f; }
  }

  s   = waveSum(s);
  sq  = waveSum(sq);
  mnp = waveMin(mnp);
  mxn = waveMax(mxn);
  pc  = waveSum(pc);
  nc  = waveSum(nc);
  if (lane == 0) {
    xbuf[0][wave] = s;  xbuf[1][wave] = sq;
    xbuf[2][wave] = mnp; xbuf[3][wave] = mxn;
    xbuf[4][wave] = pc; xbuf[5][wave] = nc;
  }
  __syncthreads();
  float rsum = 0.f, rsumsq = 0.f, min_pos = 3.0e38f, max_neg = -3.0e38f;
  float posCnt = 0.f, negCnt = 0.f;
#pragma unroll
  for (int w = 0; w < 8; ++w) {
    rsum   += xbuf[0][w];
    rsumsq += xbuf[1][w];
    min_pos = fminf(min_pos, xbuf[2][w]);
    max_neg = fmaxf(max_neg, xbuf[3][w]);
    posCnt += xbuf[4][w];
    negCnt += xbuf[5][w];
  }
  __syncthreads();

  const float mean_all  = rsum * (1.0f / (float)BN);
  const float sigma_all = rsumsq - (float)BN * mean_all * mean_all;

  // ---- phase 2: adaptive selection + exp sums (register-cached data) ----
  float sum_sel = 0.f, sq_sel = 0.f, fp_s = 0.f, fn_s = 0.f;
  float np = 0.f, nn = 0.f;
#pragma unroll
  for (int i = 0; i < 16; ++i) {
    const float v = sims[i];
    const bool same = (samemask >> i) & 1u;
    const bool pos = same && (v < 1.0f - EPSV);
    const bool neg = !same;
    const bool psel = pos && (v - MARGIN < max_neg);
    const bool nsel = neg && (v + MARGIN > min_pos);
    if (psel) {
      np += 1.f; sum_sel += v; sq_sel += v * v;
      fp_s += __expf(-(v - POS_A) / POS_B);
    }
    if (nsel) {
      nn += 1.f; sum_sel += v; sq_sel += v * v;
      fn_s += __expf((v - NEG_A) / NEG_B);
    }
  }

  sum_sel = waveSum(sum_sel);
  sq_sel  = waveSum(sq_sel);
  fp_s    = waveSum(fp_s);
  fn_s    = waveSum(fn_s);
  np      = waveSum(np);
  nn      = waveSum(nn);
  if (lane == 0) {
    xbuf[0][wave] = sum_sel; xbuf[1][wave] = sq_sel;
    xbuf[2][wave] = fp_s;    xbuf[3][wave] = fn_s;
    xbuf[4][wave] = np;      xbuf[5][wave] = nn;
  }
  __syncthreads();

  if (tid == 0) {
    float tSel = 0.f, tSelSq = 0.f, tFp = 0.f, tFn = 0.f, tNp = 0.f, tNn = 0.f;
#pragma unroll
    for (int w = 0; w < 8; ++w) {
      tSel += xbuf[0][w]; tSelSq += xbuf[1][w];
      tFp  += xbuf[2][w]; tFn    += xbuf[3][w];
      tNp  += xbuf[4][w]; tNn    += xbuf[5][w];
    }
    const float nsel = tNp + tNn;
    const float cnt  = fmaxf(nsel, 1.0f);
    const float mean_sel  = tSel / cnt;
    const float sigma_sel = (tSelSq - 2.0f * mean_sel * tSel
                             + nsel * mean_sel * mean_sel) / cnt;
    const float loss = __logf(1.0f + tFp) + __logf(1.0f + tFn)
                     + WEIGHT * (fabsf(mean_all - mean_sel)
                               + fabsf(sigma_all - sigma_sel));
    const bool valid = (posCnt > 0.f) && (negCnt > 0.f) && (tNp > 0.f) && (tNn > 0.f);
    lossrow[r] = valid ? loss : 0.0f;
  }
}

// ---------------------------------------------------------------------------
// Kernel 3: deterministic single-block final reduction (no float atomics).
// ---------------------------------------------------------------------------
__global__ __launch_bounds__(256)
void dbml_final(const float* __restrict__ lossrow, float* __restrict__ out) {
  __shared__ float sbuf[8];
  float s = 0.f;
  for (int i = threadIdx.x; i < BN; i += 256) s += lossrow[i];
  s = waveSum(s);
  if ((threadIdx.x & 31) == 0) sbuf[threadIdx.x >> 5] = s;
  __syncthreads();
  if (threadIdx.x == 0) {
    float t = 0.f;
#pragma unroll
    for (int w = 0; w < 8; ++w) t += sbuf[w];
    out[0] = t * (1.0f / (float)BN);
  }
}

extern "C" void kernel_launch(void* const* d_in, const int* in_sizes, int n_in,
                              void* d_out, int out_size, void* d_ws, size_t ws_size,
                              hipStream_t stream) {
  const float* feats  = (const float*)d_in[0];
  const int*   labels = (const int*)d_in[1];
  float* out = (float*)d_out;

  float* sim     = (float*)d_ws;                 // 4096*4096 floats = 64 MB
  float* lossrow = sim + (size_t)BN * BN;        // +16 KB

  dim3 gemmGrid(BN / 128, BN / 64);              // 32 x 64 blocks, 256 thr (8 waves)
  dbml_gemm_wmma<<<gemmGrid, 256, 0, stream>>>(feats, sim);
  dbml_rowpass<<<BN, 256, 0, stream>>>(sim, labels, lossrow);
  dbml_final<<<1, 256, 0, stream>>>(lossrow, out);
}